// GNN_47974784696843
// MI455X (gfx1250) — compile-verified
//
#include <hip/hip_runtime.h>

// ---------------------------------------------------------------------------
// GraphSAGE x3 for MI455X (gfx1250, wave32).
// out = relu(sage3(relu(sage2(relu(sage1(x))))))
// sage(h) = mean_agg(h) @ W_l^T + b_l + h @ W_r^T
//
// Bottleneck is the edge scatter (~2.4 GB of read+atomic traffic @ 23.3 TB/s);
// GEMMs (~7.5 GFLOP) run in exact fp32 via V_WMMA_F32_16X16X4_F32.
// ---------------------------------------------------------------------------

typedef float v2f __attribute__((ext_vector_type(2)));
typedef float v8f __attribute__((ext_vector_type(8)));

#define DFEAT 128

// ---------------- utility: zero a float buffer -----------------------------
__global__ void gnn_zero_f32(float* __restrict__ p, long n) {
    long i = (long)blockIdx.x * blockDim.x + threadIdx.x;
    if (i < n) p[i] = 0.0f;
}

// ---------------- degree count (fp32 atomics, exact for counts) ------------
__global__ void gnn_degree(const int* __restrict__ dst, float* __restrict__ deg,
                           int n_edges) {
    int e = blockIdx.x * blockDim.x + threadIdx.x;
    if (e < n_edges) atomicAdd(&deg[dst[e]], 1.0f);
}

__global__ void gnn_inv_deg(float* __restrict__ deg, int n_nodes) {
    int i = blockIdx.x * blockDim.x + threadIdx.x;
    if (i < n_nodes) {
        float d = deg[i];
        deg[i] = (d > 0.0f) ? (1.0f / d) : 0.0f;   // in-place: deg -> inv_deg
    }
}

// ---------------- edge scatter: agg[dst] += h[src]  (D = 128) --------------
// tid -> (edge, 16B chunk). 32 chunks/edge => consecutive lanes read/atomically
// write consecutive 16B pieces of the same feature row (fully coalesced).
__global__ void gnn_scatter_add(const float* __restrict__ h,
                                const int* __restrict__ src,
                                const int* __restrict__ dst,
                                float* __restrict__ agg,
                                int n_edges) {
    long tid = (long)blockIdx.x * blockDim.x + threadIdx.x;
    long e   = tid >> 5;            // 32 float4-chunks per 128-float row
    int  c   = (int)(tid & 31);
    if (e >= n_edges) return;
    int s = src[e], d = dst[e];
    const float4 v = *(const float4*)(h + (size_t)s * DFEAT + c * 4);
    float* o = agg + (size_t)d * DFEAT + c * 4;
    atomicAdd(o + 0, v.x);
    atomicAdd(o + 1, v.y);
    atomicAdd(o + 2, v.z);
    atomicAdd(o + 3, v.w);
}

// ---------------- fused SAGE layer GEMM (fp32 WMMA) ------------------------
// One wave computes one 16x16 output tile of:
//   out = relu( (agg * inv_deg) @ Wl^T + bl + h @ Wr^T )
// A-frag (16x4 f32): lane holds row M=lane&15, K = 2*(lane>>4)+{0,1}
// B-frag (4x16 f32): lane holds col N=lane&15 (= row of W), same K pair
// C/D   (16x16 f32): row = m0 + 8*(lane>>4) + v, col = n0 + (lane&15)
__global__ __launch_bounds__(256)
void gnn_sage_gemm_wmma(const float* __restrict__ agg,
                        const float* __restrict__ inv_deg,
                        const float* __restrict__ h,
                        const float* __restrict__ Wl,
                        const float* __restrict__ bl,
                        const float* __restrict__ Wr,
                        float* __restrict__ out,
                        int n_nodes, int d_out, int n_col_tiles) {
    const int wave = threadIdx.x >> 5;
    const int lane = threadIdx.x & 31;
    const int row_tiles_per_blk = (blockDim.x >> 5) / n_col_tiles;   // 1 or 4

    const int mtile = blockIdx.x * row_tiles_per_blk + wave / n_col_tiles;
    const int ntile = wave % n_col_tiles;
    if (mtile * 16 >= n_nodes) return;            // wave-uniform exit

    const int m0 = mtile * 16;
    const int n0 = ntile * 16;
    int rA = m0 + (lane & 15);
    if (rA >= n_nodes) rA = n_nodes - 1;          // safety clamp (M%16==0 here)
    const int rW   = n0 + (lane & 15);
    const int klo  = 2 * (lane >> 4);

    const float s = inv_deg[rA];

    const float* __restrict__ arow = agg + (size_t)rA * DFEAT + klo;
    const float* __restrict__ hrow = h   + (size_t)rA * DFEAT + klo;
    const float* __restrict__ wlr  = Wl  + (size_t)rW * DFEAT + klo;
    const float* __restrict__ wrr  = Wr  + (size_t)rW * DFEAT + klo;

    v8f acc = {};

    // mean(h) @ Wl^T : fold inv_deg scale into the A fragment
    #pragma unroll 4
    for (int kk = 0; kk < DFEAT; kk += 4) {
        v2f a = *(const v2f*)(arow + kk);
        a *= s;
        v2f b = *(const v2f*)(wlr + kk);
        acc = __builtin_amdgcn_wmma_f32_16x16x4_f32(
            false, a, false, b, (short)0, acc, false, false);
    }

    // h @ Wr^T
    #pragma unroll 4
    for (int kk = 0; kk < DFEAT; kk += 4) {
        v2f a = *(const v2f*)(hrow + kk);
        v2f b = *(const v2f*)(wrr + kk);
        acc = __builtin_amdgcn_wmma_f32_16x16x4_f32(
            false, a, false, b, (short)0, acc, false, false);
    }

    // epilogue: + bias, relu, store (C/D layout)
    const int n    = n0 + (lane & 15);
    const float bb = bl[n];
    const int mbase = m0 + 8 * (lane >> 4);
    #pragma unroll
    for (int v = 0; v < 8; ++v) {
        const int m = mbase + v;
        if (m < n_nodes) {
            float val = acc[v] + bb;
            out[(size_t)m * d_out + n] = (val > 0.0f) ? val : 0.0f;
        }
    }
}

// ---------------------------------------------------------------------------
extern "C" void kernel_launch(void* const* d_in, const int* in_sizes, int n_in,
                              void* d_out, int out_size, void* d_ws, size_t ws_size,
                              hipStream_t stream) {
    const float* x    = (const float*)d_in[0];
    const int*   eidx = (const int*)  d_in[1];     // [2, E] row-major
    const float* W_l1 = (const float*)d_in[2];
    const float* b_l1 = (const float*)d_in[3];
    const float* W_r1 = (const float*)d_in[4];
    const float* W_l2 = (const float*)d_in[5];
    const float* b_l2 = (const float*)d_in[6];
    const float* W_r2 = (const float*)d_in[7];
    const float* W_l3 = (const float*)d_in[8];
    const float* b_l3 = (const float*)d_in[9];
    const float* W_r3 = (const float*)d_in[10];
    float* out = (float*)d_out;

    const int n_nodes = in_sizes[0] / DFEAT;       // 50000
    const int n_edges = in_sizes[1] / 2;           // 800000
    const int d_out3  = out_size / n_nodes;        // 32
    const int* src = eidx;
    const int* dst = eidx + n_edges;

    // workspace layout (fp32): [deg/inv_deg N][agg N*128][h1 N*128][h2 N*128]
    float* deg = (float*)d_ws;
    float* agg = deg + n_nodes;
    float* h1  = agg + (size_t)n_nodes * DFEAT;
    float* h2  = h1  + (size_t)n_nodes * DFEAT;

    const long nfeat = (long)n_nodes * DFEAT;
    const int  TB    = 256;
    const int  gZeroN    = (int)((n_nodes + TB - 1) / TB);
    const int  gZeroFeat = (int)((nfeat + TB - 1) / TB);
    const int  gEdges    = (int)((n_edges + TB - 1) / TB);
    const int  gScatter  = (int)(((long)n_edges * 32 + TB - 1) / TB);

    const int mtiles = (n_nodes + 15) / 16;        // 3125
    // d_hid = 128 -> 8 col tiles, 1 row tile/block; d_out = 32 -> 2 col tiles, 4 row tiles/block
    const int gGemmHid = mtiles;                   // 8 waves cover 16 rows x 128 cols
    const int gGemmOut = (mtiles + 3) / 4;         // 8 waves cover 64 rows x 32 cols

    // ---- degrees -> inv_deg ----
    gnn_zero_f32<<<gZeroN, TB, 0, stream>>>(deg, n_nodes);
    gnn_degree  <<<gEdges, TB, 0, stream>>>(dst, deg, n_edges);
    gnn_inv_deg <<<gZeroN, TB, 0, stream>>>(deg, n_nodes);

    // ---- layer 1: x -> h1 ----
    gnn_zero_f32   <<<gZeroFeat, TB, 0, stream>>>(agg, nfeat);
    gnn_scatter_add<<<gScatter,  TB, 0, stream>>>(x, src, dst, agg, n_edges);
    gnn_sage_gemm_wmma<<<gGemmHid, TB, 0, stream>>>(agg, deg, x, W_l1, b_l1, W_r1,
                                                    h1, n_nodes, DFEAT, 8);

    // ---- layer 2: h1 -> h2 ----
    gnn_zero_f32   <<<gZeroFeat, TB, 0, stream>>>(agg, nfeat);
    gnn_scatter_add<<<gScatter,  TB, 0, stream>>>(h1, src, dst, agg, n_edges);
    gnn_sage_gemm_wmma<<<gGemmHid, TB, 0, stream>>>(agg, deg, h1, W_l2, b_l2, W_r2,
                                                    h2, n_nodes, DFEAT, 8);

    // ---- layer 3: h2 -> out (d_out = 32) ----
    gnn_zero_f32   <<<gZeroFeat, TB, 0, stream>>>(agg, nfeat);
    gnn_scatter_add<<<gScatter,  TB, 0, stream>>>(h2, src, dst, agg, n_edges);
    gnn_sage_gemm_wmma<<<gGemmOut, TB, 0, stream>>>(agg, deg, h2, W_l3, b_l3, W_r3,
                                                    out, n_nodes, d_out3, 2);
}